// CoAttention_76605036692245
// MI455X (gfx1250) — compile-verified
//
#include <hip/hip_runtime.h>
#include <hip/hip_bf16.h>

typedef __attribute__((ext_vector_type(2))) float v2f;
typedef __attribute__((ext_vector_type(8))) float v8f;
typedef __attribute__((ext_vector_type(4))) int   v4i;

#define Bsz 64
#define Nn  1024
#define Mm  1024
#define Dd  200
#define Kk  80
#define DP  208   // padded D (global CWl stride)
#define SP  212   // LDS stride for CWl/Sent tiles: 212%64=20 -> 16 distinct banks
#define WP  68    // LDS stride for WcC tile: 68%64=4 -> conflict-free
#define LP  68    // LDS stride for L tile

__device__ __forceinline__ v8f wmma_f32(v2f a, v2f b, v8f c) {
  // (neg_a, A, neg_b, B, c_mod, C, reuse_a, reuse_b)
  return __builtin_amdgcn_wmma_f32_16x16x4_f32(false, a, false, b, (short)0, c,
                                               false, false);
}

__device__ __forceinline__ void gatomic_fadd(float* p, float v) {
#if __has_builtin(__builtin_amdgcn_global_atomic_fadd_f32)
  typedef __attribute__((address_space(1))) float gfloat;
  __builtin_amdgcn_global_atomic_fadd_f32((gfloat*)p, v);
#else
  atomicAdd(p, v);
#endif
}

// ---- async global->LDS staging (ASYNCcnt path), with safe fallback --------
#if __has_builtin(__builtin_amdgcn_global_load_async_to_lds_b128)
#define HAVE_ASYNC_LDS 1
#endif

__device__ __forceinline__ void stage_b128(const float* g, float* l) {
#ifdef HAVE_ASYNC_LDS
  // param types per clang diagnostic: (AS1 v4i*, AS3 v4i*, imm offset, imm cpol)
  __builtin_amdgcn_global_load_async_to_lds_b128(
      (__attribute__((address_space(1))) v4i*)g,
      (__attribute__((address_space(3))) v4i*)l, 0, 0);
#else
  *(float4*)l = *(const float4*)g;
#endif
}

__device__ __forceinline__ void wait_async_stage() {
#ifdef HAVE_ASYNC_LDS
#if __has_builtin(__builtin_amdgcn_s_wait_asynccnt)
  __builtin_amdgcn_s_wait_asynccnt(0);
#else
  asm volatile("s_wait_asynccnt 0x0" ::: "memory");
#endif
#endif
}

// ---------------------------------------------------------------------------
// K1: CWl[b][m][e] = sum_d comment[b][m][d] * Wl[d][e], stored [B][M][DP]
// ---------------------------------------------------------------------------
__global__ __launch_bounds__(256) void k_cwl(const float* __restrict__ comm,
                                             const float* __restrict__ Wl,
                                             float* __restrict__ CWl) {
  const int b    = blockIdx.y;
  const int lane = threadIdx.x & 31;
  const int wv   = threadIdx.x >> 5;
  const int t    = blockIdx.x * 8 + wv;          // 0..831
  const int m0   = (t / 13) * 16;
  const int e0   = (t % 13) * 16;
  const int lid  = lane & 15;
  const int half = lane >> 4;
  const int kb   = half * 2;

  const float* arow = comm + ((size_t)b * Mm + (m0 + lid)) * Dd;
  const int e       = e0 + lid;
  const int ec      = (e < Dd) ? e : (Dd - 1);
  const float msk   = (e < Dd) ? 1.f : 0.f;

  v8f c = {};
  for (int d0 = 0; d0 < Dd; d0 += 4) {
    v2f a = *(const v2f*)(arow + d0 + kb);
    v2f bb;
    bb.x = Wl[(d0 + kb) * Dd + ec] * msk;
    bb.y = Wl[(d0 + kb + 1) * Dd + ec] * msk;
    c = wmma_f32(a, bb, c);
  }
  float* orow = CWl + (size_t)b * Mm * DP;
#pragma unroll
  for (int v = 0; v < 8; ++v)
    orow[(size_t)(m0 + v + half * 8) * DP + e] = c[v];
}

// ---------------------------------------------------------------------------
// K2: out[b][k][c] = sum_d W[k][d] * X[b][c][d]
// ---------------------------------------------------------------------------
__global__ __launch_bounds__(256) void k_proj(const float* __restrict__ W,
                                              const float* __restrict__ X,
                                              float* __restrict__ out) {
  const int b    = blockIdx.y;
  const int lane = threadIdx.x & 31;
  const int wv   = threadIdx.x >> 5;
  const int t    = blockIdx.x * 8 + wv;          // 0..319
  const int k0   = (t >> 6) * 16;
  const int c0   = (t & 63) * 16;
  const int lid  = lane & 15;
  const int half = lane >> 4;
  const int kb   = half * 2;

  const float* arow = W + (size_t)(k0 + lid) * Dd;
  const float* brow = X + ((size_t)b * Nn + (c0 + lid)) * Dd;

  v8f c = {};
  for (int d0 = 0; d0 < Dd; d0 += 4) {
    v2f a  = *(const v2f*)(arow + d0 + kb);
    v2f bb = *(const v2f*)(brow + d0 + kb);
    c = wmma_f32(a, bb, c);
  }
#pragma unroll
  for (int v = 0; v < 8; ++v)
    out[((size_t)b * Kk + k0 + v + half * 8) * Nn + c0 + lid] = c[v];
}

// ---------------------------------------------------------------------------
// K3: fused L / T1 / T2 kernel (block = (b, 64-row m-tile); loops 64-wide n)
// ---------------------------------------------------------------------------
__global__ __launch_bounds__(256) void k_fused(const float* __restrict__ sent,
                                               const float* __restrict__ CWl,
                                               const float* __restrict__ WsS,
                                               const float* __restrict__ WcC,
                                               float* __restrict__ T1,
                                               float* __restrict__ T2) {
  extern __shared__ float smem[];
  float* sCWl  = smem;                 // 64*SP
  float* sSent = sCWl + 64 * SP;       // 64*SP
  float* sWcC  = sSent + 64 * SP;      // 80*WP
  float* sL    = sWcC + Kk * WP;       // 64*LP

  const int b    = blockIdx.y;
  const int m0   = blockIdx.x * 64;
  const int tid  = threadIdx.x;
  const int wv   = tid >> 5;
  const int lane = tid & 31;
  const int lid  = lane & 15;
  const int half = lane >> 4;
  const int kb   = half * 2;

  // stage CWl rows [m0, m0+64): 52 float4 per row, async
  for (int i = tid; i < 64 * (DP / 4); i += 256) {
    const int r = i / (DP / 4), c4 = (i % (DP / 4)) * 4;
    stage_b128(CWl + ((size_t)b * Mm + m0 + r) * DP + c4, sCWl + r * SP + c4);
  }
  // stage WcC[:, m0:m0+64] with stride WP
  for (int i = tid; i < Kk * 64; i += 256)
    sWcC[(i >> 6) * WP + (i & 63)] =
        WcC[((size_t)b * Kk + (i >> 6)) * Mm + m0 + (i & 63)];

  v8f acc2[3] = {};

  const int mi0 = (wv >> 2) * 16, ni0 = (wv & 3) * 16;   // L subtile 0
  const int mi1 = mi0 + 32;                              // L subtile 1

  for (int n0 = 0; n0 < Nn; n0 += 64) {
    wait_async_stage();
    __syncthreads();  // previous iteration readers of sSent/sL done
    // stage sentence rows [n0, n0+64): 50 float4 per row, async
    for (int i = tid; i < 64 * (Dd / 4); i += 256) {
      const int r = i / (Dd / 4), c4 = (i % (Dd / 4)) * 4;
      stage_b128(sent + ((size_t)b * Nn + n0 + r) * Dd + c4,
                 sSent + r * SP + c4);
    }
    // prefetch next n-tile into L2 while this one computes
    if (n0 + 64 < Nn) {
      for (int i = tid; i < 64 * 7; i += 256) {
        const int r = i / 7, c = (i % 7) * 32;
        __builtin_prefetch(sent + ((size_t)b * Nn + n0 + 64 + r) * Dd + c, 0, 1);
      }
    }
    wait_async_stage();
    __syncthreads();

    // ---- L tile (64x64): each wave computes two 16x16 subtiles ----
    v8f cl0 = {}, cl1 = {};
    const float* a0p = sCWl + (mi0 + lid) * SP;
    const float* a1p = sCWl + (mi1 + lid) * SP;
    const float* bp  = sSent + (ni0 + lid) * SP;
    for (int e = 0; e < Dd; e += 4) {
      v2f a0 = *(const v2f*)(a0p + e + kb);
      v2f a1 = *(const v2f*)(a1p + e + kb);
      v2f bb = *(const v2f*)(bp + e + kb);
      cl0 = wmma_f32(a0, bb, cl0);
      cl1 = wmma_f32(a1, bb, cl1);
    }
#pragma unroll
    for (int v = 0; v < 8; ++v) {
      sL[(mi0 + v + half * 8) * LP + ni0 + lid] = tanhf(cl0[v]);
      sL[(mi1 + v + half * 8) * LP + ni0 + lid] = tanhf(cl1[v]);
    }
    __syncthreads();

    // ---- T1 partial (atomic) + T2 accumulate: 20 tiles, <=3 per wave ----
#pragma unroll
    for (int it = 0; it < 3; ++it) {
      const int t = wv + it * 8;
      if (t < 20) {                       // wave-uniform guard
        const int k0t  = (t >> 2) * 16;
        const int cl16 = (t & 3) * 16;

        // T1: A = sWcC[k][mm], B[mm][n] = sL[mm][n]
        v8f p = {};
        const float* wa = sWcC + (k0t + lid) * WP;
        for (int mm = 0; mm < 64; mm += 4) {
          v2f a = *(const v2f*)(wa + mm + kb);
          v2f bb;
          bb.x = sL[(mm + kb) * LP + cl16 + lid];
          bb.y = sL[(mm + kb + 1) * LP + cl16 + lid];
          p = wmma_f32(a, bb, p);
        }
        float* t1p =
            T1 + ((size_t)b * Kk + k0t + half * 8) * Nn + n0 + cl16 + lid;
#pragma unroll
        for (int v = 0; v < 8; ++v) gatomic_fadd(t1p + (size_t)v * Nn, p[v]);

        // T2: A = WsS[b][k][n0+nn], B[nn][m] = sL[m][nn]
        const float* wsa = WsS + ((size_t)b * Kk + k0t + lid) * Nn + n0;
        const float* slb = sL + (cl16 + lid) * LP;
        v8f q = acc2[it];
        for (int nn = 0; nn < 64; nn += 4) {
          v2f a  = *(const v2f*)(wsa + nn + kb);
          v2f bb = *(const v2f*)(slb + nn + kb);
          q = wmma_f32(a, bb, q);
        }
        acc2[it] = q;
      }
    }
  }

  // write T2 (this block owns the m-tile: plain stores)
#pragma unroll
  for (int it = 0; it < 3; ++it) {
    const int t = wv + it * 8;
    if (t < 20) {
      const int k0t = (t >> 2) * 16;
      const int ml  = (t & 3) * 16;
      float* t2p =
          T2 + ((size_t)b * Kk + k0t + half * 8) * Mm + m0 + ml + lid;
#pragma unroll
      for (int v = 0; v < 8; ++v) t2p[(size_t)v * Mm] = acc2[it][v];
    }
  }
}

// ---------------------------------------------------------------------------
// K4: ss[b][n] = sum_k wh[k] * tanh(P[b][k][n] + T[b][k][n])
// ---------------------------------------------------------------------------
__global__ __launch_bounds__(256) void k_logits(const float* __restrict__ P,
                                                const float* __restrict__ T,
                                                const float* __restrict__ wh,
                                                float* __restrict__ ss) {
  const int b = blockIdx.y;
  const int n = blockIdx.x * 256 + threadIdx.x;
  const float* pp = P + (size_t)b * Kk * Nn + n;
  const float* tt = T + (size_t)b * Kk * Nn + n;
  float acc = 0.f;
  for (int k = 0; k < Kk; ++k)
    acc += wh[k] * tanhf(pp[(size_t)k * Nn] + tt[(size_t)k * Nn]);
  ss[(size_t)b * Nn + n] = acc;
}

// ---------------------------------------------------------------------------
// K5: softmax over the BATCH axis (legacy dim=0), in place on ss[B][1024]
// ---------------------------------------------------------------------------
__global__ __launch_bounds__(256) void k_softmax_b(float* __restrict__ ss) {
  const int j = blockIdx.x * 256 + threadIdx.x;
  float mx = -3.4e38f;
  for (int b = 0; b < Bsz; ++b) mx = fmaxf(mx, ss[(size_t)b * Nn + j]);
  float s = 0.f;
  for (int b = 0; b < Bsz; ++b) s += expf(ss[(size_t)b * Nn + j] - mx);
  const float inv = 1.f / s;
  for (int b = 0; b < Bsz; ++b)
    ss[(size_t)b * Nn + j] = expf(ss[(size_t)b * Nn + j] - mx) * inv;
}

// ---------------------------------------------------------------------------
// K6: out[b][off+d] = sum_n rep[b][n][d] * A[b][n]
// ---------------------------------------------------------------------------
__global__ __launch_bounds__(256) void k_contract(const float* __restrict__ rep,
                                                  const float* __restrict__ A,
                                                  float* __restrict__ out,
                                                  int off) {
  const int b = blockIdx.x;
  const int d = threadIdx.x;
  if (d >= Dd) return;
  const float* r = rep + (size_t)b * Nn * Dd + d;
  const float* a = A + (size_t)b * Nn;
  float acc = 0.f;
  for (int n = 0; n < Nn; ++n) acc += r[(size_t)n * Dd] * a[n];
  out[b * (2 * Dd) + off + d] = acc;
}

// ---------------------------------------------------------------------------
extern "C" void kernel_launch(void* const* d_in, const int* in_sizes, int n_in,
                              void* d_out, int out_size, void* d_ws,
                              size_t ws_size, hipStream_t stream) {
  const float* sent = (const float*)d_in[0];
  const float* comm = (const float*)d_in[1];
  const float* Wl   = (const float*)d_in[2];
  const float* Wc   = (const float*)d_in[3];
  const float* Ws   = (const float*)d_in[4];
  const float* whs  = (const float*)d_in[5];
  const float* whc  = (const float*)d_in[6];
  float* out = (float*)d_out;
  float* ws  = (float*)d_ws;

  float* CWl = ws;                                   // B*M*DP
  float* WsS = CWl + (size_t)Bsz * Mm * DP;          // B*K*N
  float* WcC = WsS + (size_t)Bsz * Kk * Nn;          // B*K*M
  float* T1  = WcC + (size_t)Bsz * Kk * Mm;          // B*K*N
  float* T2  = T1 + (size_t)Bsz * Kk * Nn;           // B*K*M
  float* SS  = T2 + (size_t)Bsz * Kk * Mm;           // 2*B*1024

  (void)hipMemsetAsync(T1, 0, (size_t)Bsz * Kk * Nn * sizeof(float), stream);

  k_cwl<<<dim3(104, Bsz), 256, 0, stream>>>(comm, Wl, CWl);
  k_proj<<<dim3(40, Bsz), 256, 0, stream>>>(Ws, sent, WsS);
  k_proj<<<dim3(40, Bsz), 256, 0, stream>>>(Wc, comm, WcC);

  const size_t lds =
      (size_t)(64 * SP + 64 * SP + Kk * WP + 64 * LP) * sizeof(float);
  k_fused<<<dim3(Mm / 64, Bsz), 256, lds, stream>>>(sent, CWl, WsS, WcC, T1,
                                                    T2);

  k_logits<<<dim3(4, Bsz), 256, 0, stream>>>(WsS, T1, whs, SS);
  k_logits<<<dim3(4, Bsz), 256, 0, stream>>>(WcC, T2, whc, SS + (size_t)Bsz * Nn);
  k_softmax_b<<<dim3(4), 256, 0, stream>>>(SS);
  k_softmax_b<<<dim3(4), 256, 0, stream>>>(SS + (size_t)Bsz * Nn);
  k_contract<<<dim3(Bsz), 256, 0, stream>>>(sent, SS, out, 0);
  k_contract<<<dim3(Bsz), 256, 0, stream>>>(comm, SS + (size_t)Bsz * Nn, out, Dd);
}